// RoIPool_49357764165672
// MI455X (gfx1250) — compile-verified
//
#include <hip/hip_runtime.h>
#include <cstdint>

#define POOLED 7
#define GS 7
#define D_OUT 21
#define N_BATCH 2
#define CHANS (D_OUT * GS * GS)          // 1029
#define H_F 96
#define W_F 96
#define PLANE (H_F * W_F)                // 9216
#define N_PLANES (N_BATCH * CHANS)       // 2058
#define R_ROIS 1024
#define SCALE 0.0625f
#define TOTAL_OUT (R_ROIS * D_OUT * POOLED * POOLED)   // 1,053,696 = 4116*256

typedef __attribute__((ext_vector_type(2))) float    v2f;
typedef __attribute__((ext_vector_type(8))) float    v8f;
typedef __attribute__((ext_vector_type(4))) unsigned v4u;
typedef __attribute__((ext_vector_type(8))) int      v8i;
typedef __attribute__((ext_vector_type(4))) int      v4i;

// ---------------------------------------------------------------------------
// Kernel A: per-(n,c) plane 2-D inclusive prefix sum (summed-area table).
// One wave32 per plane. Plane is staged global->LDS via the Tensor Data Mover,
// scanned with v_wmma_f32_16x16x4_f32 (prefix-sum == multiply by triangular
// ones matrix), and written back LDS->global via TDM.
// ---------------------------------------------------------------------------
__global__ __launch_bounds__(32) void RoIPool_sat_kernel(
    const float* __restrict__ feat, float* __restrict__ sat) {
  __shared__ float tile[PLANE];

  const int lane = threadIdx.x;      // 0..31
  const int half = lane >> 4;        // 0 or 1 (16-lane halves)
  const int l    = lane & 15;
  const int plane = blockIdx.x;

  // ---- TDM async tensor load: 96x96 f32 tile, global -> LDS ----
  {
    uint64_t ga = (uint64_t)(uintptr_t)(feat + (size_t)plane * PLANE);
    uint32_t la = (uint32_t)(uintptr_t)tile;   // LDS byte offset (flat addr low 32b)
    v4u g0 = { 1u,                                   // count=1 (valid descriptor)
               la,                                   // lds_addr
               (uint32_t)ga,                         // global_addr[31:0]
               (uint32_t)((ga >> 32) & 0x1FFFFFFu) | (2u << 30) };  // addr[56:32] | type=2
    v8i g1 = { (int)(2u << 16),     // data_size = 4B
               (int)(96u << 16),    // tensor_dim0 = 96
               (int)(96u << 16),    // tensor_dim1 = 96
               (int)(96u << 16),    // tile_dim0   = 96
               96,                  // tile_dim1   = 96 (tile_dim2 = 0)
               96,                  // tensor_dim0_stride = 96
               0, 0 };
    v4i gz4 = { 0, 0, 0, 0 };
    v8i gz8 = { 0, 0, 0, 0, 0, 0, 0, 0 };
    __builtin_amdgcn_tensor_load_to_lds(g0, g1, gz4, gz4, gz8, 0);
    __builtin_amdgcn_s_wait_tensorcnt(0);
    asm volatile("" ::: "memory");
  }

  // ---- Phase 1: inclusive scan along W.  Y = X * U  (U[i][j] = i<=j) ----
  // A (16x4 f32) layout: v0 = {K=0 | K=2}, v1 = {K=1 | K=3}, M = lane%16.
  // B (4x16 f32) layout: v0 = {row K=0 | row K=2}, v1 = {K=1 | K=3}, N = lane%16.
  // C/D (16x16 f32): VGPR v holds M=v (lanes 0-15) / M=v+8 (lanes 16-31), N = lane%16.
  for (int rb = 0; rb < 6; ++rb) {
    v8f carry = {};
    for (int cb = 0; cb < 6; ++cb) {
      v8f acc = carry;
#pragma unroll
      for (int kk = 0; kk < 16; kk += 4) {
        v2f a, b;
        const int col0 = cb * 16 + kk + 2 * half;
        a.x = tile[(rb * 16 + l) * W_F + col0];
        a.y = tile[(rb * 16 + l) * W_F + col0 + 1];
        b.x = ((kk + 2 * half)     <= l) ? 1.0f : 0.0f;   // U[kk+2h][l]
        b.y = ((kk + 2 * half + 1) <= l) ? 1.0f : 0.0f;
        acc = __builtin_amdgcn_wmma_f32_16x16x4_f32(false, a, false, b,
                                                    (short)0, acc, false, false);
      }
#pragma unroll
      for (int v = 0; v < 8; ++v)
        tile[(rb * 16 + v + 8 * half) * W_F + cb * 16 + l] = acc[v];
      // carry[r] = Y[r][15]: column 15 lives in lane 15 / 31 of each VGPR.
#pragma unroll
      for (int v = 0; v < 8; ++v)
        carry[v] = __shfl(acc[v], 15, 16);
    }
  }
  __syncthreads();

  // ---- Phase 2: inclusive scan along H.  Z = L * Y  (L[i][j] = j<=i) ----
  for (int cb = 0; cb < 6; ++cb) {
    float carry = 0.0f;                       // carry[N], broadcast per lane (N = l)
    for (int rb = 0; rb < 6; ++rb) {
      v8f acc;
#pragma unroll
      for (int v = 0; v < 8; ++v) acc[v] = carry;
#pragma unroll
      for (int kk = 0; kk < 16; kk += 4) {
        v2f a, b;
        a.x = ((kk + 2 * half)     <= l) ? 1.0f : 0.0f;   // L[l][kk+2h]
        a.y = ((kk + 2 * half + 1) <= l) ? 1.0f : 0.0f;
        const int row0 = rb * 16 + kk + 2 * half;
        b.x = tile[row0 * W_F + cb * 16 + l];
        b.y = tile[(row0 + 1) * W_F + cb * 16 + l];
        acc = __builtin_amdgcn_wmma_f32_16x16x4_f32(false, a, false, b,
                                                    (short)0, acc, false, false);
      }
#pragma unroll
      for (int v = 0; v < 8; ++v)
        tile[(rb * 16 + v + 8 * half) * W_F + cb * 16 + l] = acc[v];
      // carry[N] = Z[15][N]: row M=15 = VGPR7, lanes 16..31.
      carry = __shfl(acc[7], 16 + l, 32);
    }
  }

  // ---- TDM async tensor store: LDS -> SAT plane in workspace ----
  asm volatile("s_wait_dscnt 0x0" ::: "memory");   // LDS writes visible to TDM
  {
    uint64_t ga = (uint64_t)(uintptr_t)(sat + (size_t)plane * PLANE);
    uint32_t la = (uint32_t)(uintptr_t)tile;
    v4u g0 = { 1u, la, (uint32_t)ga,
               (uint32_t)((ga >> 32) & 0x1FFFFFFu) | (2u << 30) };
    v8i g1 = { (int)(2u << 16), (int)(96u << 16), (int)(96u << 16),
               (int)(96u << 16), 96, 96, 0, 0 };
    v4i gz4 = { 0, 0, 0, 0 };
    v8i gz8 = { 0, 0, 0, 0, 0, 0, 0, 0 };
    __builtin_amdgcn_tensor_store_from_lds(g0, g1, gz4, gz4, gz8, 0);
  }
  // implicit s_wait_idle at s_endpgm covers TENSORcnt
}

// ---------------------------------------------------------------------------
// Bin-bound decode shared by both consumers. Matches the reference bit-for-bit
// where it matters: rintf == jnp.round (half-to-even), unfused mul/add, clamp
// in float to [0,96] then cast.
// ---------------------------------------------------------------------------
__device__ __forceinline__ void decode_bin(
    const float* __restrict__ rois, int r, int d, int ph, int pw,
    int& b, int& c, int& h0, int& h1, int& w0, int& w1) {
  const float* roi = rois + r * 5;
  b = (int)roi[0];
  float xs = rintf(roi[1]) * SCALE;
  float ys = rintf(roi[2]) * SCALE;
  float xe = (rintf(roi[3]) + 1.0f) * SCALE;
  float ye = (rintf(roi[4]) + 1.0f) * SCALE;
  float bw = fmaxf(xe - xs, 0.1f) / 7.0f;
  float bh = fmaxf(ye - ys, 0.1f) / 7.0f;
  h0 = (int)fminf(fmaxf(floorf(__fadd_rn(__fmul_rn((float)ph, bh), ys)), 0.f), 96.f);
  h1 = (int)fminf(fmaxf(ceilf (__fadd_rn(__fmul_rn((float)(ph + 1), bh), ys)), 0.f), 96.f);
  w0 = (int)fminf(fmaxf(floorf(__fadd_rn(__fmul_rn((float)pw, bw), xs)), 0.f), 96.f);
  w1 = (int)fminf(fmaxf(ceilf (__fadd_rn(__fmul_rn((float)(pw + 1), bw), xs)), 0.f), 96.f);
  c = (d * GS + ph) * GS + pw;
}

// ---------------------------------------------------------------------------
// Kernel B: one thread per output; 4 SAT corner lookups per bin.
// ---------------------------------------------------------------------------
__global__ __launch_bounds__(256) void RoIPool_gather_kernel(
    const float* __restrict__ sat, const float* __restrict__ rois,
    float* __restrict__ out) {
  int t = blockIdx.x * 256 + threadIdx.x;
  if (t >= TOTAL_OUT) return;
  int pw = t % POOLED;
  int tmp = t / POOLED;
  int ph = tmp % POOLED;  tmp /= POOLED;
  int d  = tmp % D_OUT;
  int r  = tmp / D_OUT;

  int b, c, h0, h1, w0, w1;
  decode_bin(rois, r, d, ph, pw, b, c, h0, h1, w0, w1);

  const float* P = sat + ((size_t)(b * CHANS + c)) * PLANE;
  auto Ip = [&](int h, int w) -> float {
    return (h == 0 || w == 0) ? 0.0f : P[(h - 1) * W_F + (w - 1)];
  };
  float S = Ip(h1, w1) - Ip(h0, w1) - Ip(h1, w0) + Ip(h0, w0);
  int area = (h1 - h0) * (w1 - w0);
  out[t] = (area > 0) ? S / (float)area : 0.0f;
}

// ---------------------------------------------------------------------------
// Fallback: direct summation (used only if workspace can't hold the SAT).
// ---------------------------------------------------------------------------
__global__ __launch_bounds__(256) void RoIPool_direct_kernel(
    const float* __restrict__ feat, const float* __restrict__ rois,
    float* __restrict__ out) {
  int t = blockIdx.x * 256 + threadIdx.x;
  if (t >= TOTAL_OUT) return;
  int pw = t % POOLED;
  int tmp = t / POOLED;
  int ph = tmp % POOLED;  tmp /= POOLED;
  int d  = tmp % D_OUT;
  int r  = tmp / D_OUT;

  int b, c, h0, h1, w0, w1;
  decode_bin(rois, r, d, ph, pw, b, c, h0, h1, w0, w1);

  const float* P = feat + ((size_t)(b * CHANS + c)) * PLANE;
  float sum = 0.0f;
  for (int h = h0; h < h1; ++h) {
    const float* row = P + h * W_F;
    for (int w = w0; w < w1; ++w) sum += row[w];
  }
  int area = (h1 - h0) * (w1 - w0);
  out[t] = (area > 0) ? sum / (float)area : 0.0f;
}

extern "C" void kernel_launch(void* const* d_in, const int* in_sizes, int n_in,
                              void* d_out, int out_size, void* d_ws, size_t ws_size,
                              hipStream_t stream) {
  (void)in_sizes; (void)n_in; (void)out_size;
  const float* feat = (const float*)d_in[0];
  const float* rois = (const float*)d_in[1];
  float* out = (float*)d_out;

  const size_t satBytes = (size_t)N_PLANES * PLANE * sizeof(float);  // 75.9 MB
  if (ws_size >= satBytes) {
    RoIPool_sat_kernel<<<dim3(N_PLANES), dim3(32), 0, stream>>>(feat, (float*)d_ws);
    RoIPool_gather_kernel<<<dim3(TOTAL_OUT / 256), dim3(256), 0, stream>>>(
        (const float*)d_ws, rois, out);
  } else {
    RoIPool_direct_kernel<<<dim3(TOTAL_OUT / 256), dim3(256), 0, stream>>>(
        feat, rois, out);
  }
}